// QLSTM_65481071397828
// MI455X (gfx1250) — compile-verified
//
#include <hip/hip_runtime.h>
#include <math.h>

typedef float v2f __attribute__((ext_vector_type(2)));
typedef float v8f __attribute__((ext_vector_type(8)));
typedef int   v4i __attribute__((vector_size(16)));
typedef __attribute__((address_space(1))) v4i* gv4i_p;   // global (device) ptr
typedef __attribute__((address_space(3))) v4i* sv4i_p;   // LDS ptr

#define T_   512
#define B_   64
#define D_   256
#define H_   512
#define NQ_  64
#define GC_  256          // 4 gates * NQ
#define NWG_ 8
#define QS_  32           // q columns owned per WG (GC_/NWG_)
#define HS_  64           // H columns owned per WG (H_/NWG_)
#define HP_  516          // LDS pitch for h rows   (bank-conflict-free)
#define WHP_ 40           // LDS pitch for Wh rows  (2*40 mod 64 = 16)
#define MP_  72           // LDS pitch for M rows   (2*72 mod 64 = 16)

// CDNA5 async global->LDS copy (ASYNCcnt path), if the toolchain exposes it.
#if defined(__has_builtin)
#if __has_builtin(__builtin_amdgcn_global_load_async_to_lds_b128)
#define HAVE_ASYNC_LDS 1
#endif
#endif
#ifndef HAVE_ASYNC_LDS
#define HAVE_ASYNC_LDS 0
#endif

__device__ __forceinline__ v8f wmma_f32(v2f a, v2f b, v8f c) {
  // V_WMMA_F32_16X16X4_F32 : D = A(16x4) * B(4x16) + C(16x16), fp32
  return __builtin_amdgcn_wmma_f32_16x16x4_f32(
      false, a, false, b, (short)0, c, false, false);
}

// ---------------------------------------------------------------------------
// Kernel 0: zero the grid barrier state (d_ws is poisoned by the harness)
// ---------------------------------------------------------------------------
__global__ void qlstm_init(unsigned* bar) {
  bar[0] = 0u;  // arrival count
  bar[1] = 0u;  // generation
}

// ---------------------------------------------------------------------------
// Kernel 1: xz[t*B+b, :] = x[t,b,:] @ Wx_all + bias   (fully parallel GEMM)
//   A: x reshaped [32768, 256]   B: gate-concat Wx [256, 256]
//   One 16x16 C-tile per wave, K-loop of 64 fp32 WMMAs.
// ---------------------------------------------------------------------------
__global__ __launch_bounds__(256) void qlstm_xz(
    const float* __restrict__ x,
    const float* __restrict__ Wf, const float* __restrict__ bf,
    const float* __restrict__ Wi, const float* __restrict__ bi,
    const float* __restrict__ Wg, const float* __restrict__ bg,
    const float* __restrict__ Wo, const float* __restrict__ bo,
    float* __restrict__ xz)
{
  const int lane = threadIdx.x & 31;
  const int wave = threadIdx.x >> 5;
  const int half = lane >> 4;
  const int l16  = lane & 15;

  const int tile = blockIdx.x * 8 + wave;     // 0..32767
  const int m0   = (tile >> 4) << 4;          // row tile base in [0, 32768)
  const int n0   = (tile & 15) << 4;          // col tile base in [0, 256)
  const int g    = n0 >> 6;                   // whole tile lies in one gate
  const float* W  = (g == 0) ? Wf : (g == 1) ? Wi : (g == 2) ? Wg : Wo;
  const float* bb = (g == 0) ? bf : (g == 1) ? bi : (g == 2) ? bg : bo;
  const int nq = (n0 & 63) + l16;

  v8f acc = {};
  const float* arow = x + (size_t)(m0 + l16) * D_ + 2 * half;
  const float* bcol = W + nq + (size_t)(2 * half) * NQ_;
  for (int k = 0; k < D_; k += 4) {
    v2f a = *(const v2f*)(arow + k);
    v2f b;
    b.x = bcol[(size_t)k * NQ_];
    b.y = bcol[(size_t)k * NQ_ + NQ_];
    acc = wmma_f32(a, b, acc);
  }
  const float bias = bb[nq];
#pragma unroll
  for (int r = 0; r < 8; ++r) {
    const int row = m0 + r + 8 * half;
    xz[(size_t)row * GC_ + n0 + l16] = acc[r] + bias;
  }
}

// ---------------------------------------------------------------------------
// Grid barrier across the 8 resident workgroups (sense via generation count)
// ---------------------------------------------------------------------------
__device__ __forceinline__ void grid_barrier(unsigned* count, unsigned* gen) {
  __syncthreads();
  if (threadIdx.x == 0) {
    __threadfence();
    unsigned g = atomicAdd(gen, 0u);                 // read current generation
    if (atomicAdd(count, 1u) == NWG_ - 1) {          // last to arrive
      *count = 0u;
      __threadfence();
      atomicAdd(gen, 1u);                            // release
    } else {
      while (atomicAdd(gen, 0u) == g)
        __builtin_amdgcn_s_sleep(2);
    }
    __threadfence();
  }
  __syncthreads();
}

// ---------------------------------------------------------------------------
// Kernel 2: the recurrence. 8 WGs x 256 threads (8 wave32 each).
//   WG w owns q-cols [w*32, w*32+32) and H-cols [w*64, w*64+64).
//   LDS: Wh slice [512][40-pitch], M slices [4][64][72-pitch], h [64][516-pitch]
// ---------------------------------------------------------------------------
__global__ __launch_bounds__(256) void qlstm_rec(
    const float* __restrict__ Wf, const float* __restrict__ Wi,
    const float* __restrict__ Wgt, const float* __restrict__ Wo,
    const float* __restrict__ Mf, const float* __restrict__ mbf,
    const float* __restrict__ Mi, const float* __restrict__ mbi,
    const float* __restrict__ Mg, const float* __restrict__ mbg,
    const float* __restrict__ Mo, const float* __restrict__ mbo,
    const float* __restrict__ xz,
    float* __restrict__ out,      // [T][B][H] then hT[B][H] then cT[B][H]
    float* __restrict__ hbuf,     // [B][H]   global h state
    float* __restrict__ qbuf,     // [B][GC]  global q staging
    unsigned* __restrict__ bar)
{
  extern __shared__ float smem[];
  float* sWh = smem;                       // [512][WHP_]
  float* sM  = sWh + 512 * WHP_;           // [4][64][MP_]
  float* sH  = sM + 4 * 64 * MP_;          // [B_][HP_]

  const int tid  = threadIdx.x;
  const int wg   = blockIdx.x;
  const int lane = tid & 31;
  const int wave = tid >> 5;
  const int half = lane >> 4;
  const int l16  = lane & 15;

  const float* Warr[4]  = {Wf, Wi, Wgt, Wo};
  const float* Marr[4]  = {Mf, Mi, Mg, Mo};
  const float* mbarr[4] = {mbf, mbi, mbg, mbo};

  // ---- preload weight slices into LDS ----
  for (int idx = tid; idx < 512 * QS_; idx += 256) {
    const int k = idx / QS_, j = idx % QS_;
    const int c = wg * QS_ + j, g = c >> 6, nq = c & 63;
    sWh[k * WHP_ + j] = Warr[g][(size_t)(D_ + k) * NQ_ + nq];
  }
  for (int idx = tid; idx < 4 * 64 * HS_; idx += 256) {
    const int g = idx >> 12, r = (idx >> 6) & 63, cc = idx & 63;
    sM[g * (64 * MP_) + r * MP_ + cc] = Marr[g][(size_t)r * H_ + wg * HS_ + cc];
  }
  // ---- zero our slice of the global h state ----
  for (int idx = tid; idx < B_ * HS_; idx += 256) {
    const int row = idx / HS_, col = idx % HS_;
    hbuf[row * H_ + wg * HS_ + col] = 0.f;
  }
  __syncthreads();
  grid_barrier(bar, bar + 1);

  // tile assignment
  const int am0 = (wave >> 1) << 4;        // phase A: row tile
  const int an0 = (wave & 1) << 4;         // phase A: col tile within QS_
  const int bm0 = (wave >> 1) << 4;        // phase B: row tile
  const int bn0 = (wave & 1) * 32;         // phase B: cols {bn0, bn0+16}

  v8f cst[2] = {{}, {}};                   // cell-state fragments (persist)

  for (int t = 0; t < T_; ++t) {
    // ---- stage full h into LDS ----
#if HAVE_ASYNC_LDS
    {
      for (int idx = tid; idx < B_ * H_ / 4; idx += 256) {
        const int row = idx >> 7, c4 = idx & 127;
        __builtin_amdgcn_global_load_async_to_lds_b128(
            (gv4i_p)(v4i*)(hbuf + (size_t)idx * 4),
            (sv4i_p)(v4i*)&sH[row * HP_ + c4 * 4],
            0, 0);
      }
      asm volatile("s_wait_asynccnt 0" ::: "memory");
    }
#else
    {
      const float4* src = (const float4*)hbuf;
      for (int idx = tid; idx < B_ * H_ / 4; idx += 256) {
        const int row = idx >> 7, c4 = idx & 127;
        *(float4*)&sH[row * HP_ + c4 * 4] = src[idx];
      }
    }
#endif
    __syncthreads();

    // ---- Phase A: q-slice = h @ Wh_slice + xz[t]  ->  qbuf ----
    {
      v8f acc = {};
      const float* ap = &sH[(am0 + l16) * HP_ + 2 * half];
      const float* bp = &sWh[(2 * half) * WHP_ + an0 + l16];
      for (int k = 0; k < H_; k += 4) {
        v2f a = *(const v2f*)(ap + k);
        v2f b;
        b.x = bp[k * WHP_];
        b.y = bp[k * WHP_ + WHP_];
        acc = wmma_f32(a, b, acc);
      }
      const int coln = wg * QS_ + an0 + l16;
#pragma unroll
      for (int r = 0; r < 8; ++r) {
        const int row = am0 + r + 8 * half;
        const float v = acc[r] + xz[((size_t)t * B_ + row) * GC_ + coln];
        qbuf[row * GC_ + coln] = v;
      }
    }
    grid_barrier(bar, bar + 1);

    // ---- Phase B: s_g = q_g @ M_g + mb_g ; LSTM update ; write h ----
    {
      v8f s[4][2];
#pragma unroll
      for (int g = 0; g < 4; ++g) {
        v8f a0 = {}, a1 = {};
        const float* qp  = qbuf + (size_t)(bm0 + l16) * GC_ + g * NQ_ + 2 * half;
        const float* mp0 = &sM[g * (64 * MP_) + (2 * half) * MP_ + bn0 + l16];
        const float* mp1 = mp0 + 16;
        for (int k = 0; k < NQ_; k += 4) {
          v2f a = *(const v2f*)(qp + k);
          v2f b0, b1;
          b0.x = mp0[k * MP_]; b0.y = mp0[k * MP_ + MP_];
          b1.x = mp1[k * MP_]; b1.y = mp1[k * MP_ + MP_];
          a0 = wmma_f32(a, b0, a0);
          a1 = wmma_f32(a, b1, a1);
        }
        const float mb0 = mbarr[g][wg * HS_ + bn0 + l16];
        const float mb1 = mbarr[g][wg * HS_ + bn0 + 16 + l16];
#pragma unroll
        for (int r = 0; r < 8; ++r) { a0[r] += mb0; a1[r] += mb1; }
        s[g][0] = a0; s[g][1] = a1;
      }
#pragma unroll
      for (int j = 0; j < 2; ++j) {
        const int coln = wg * HS_ + bn0 + j * 16 + l16;
#pragma unroll
        for (int r = 0; r < 8; ++r) {
          const float fv = 1.f / (1.f + __expf(-s[0][j][r]));
          const float iv = 1.f / (1.f + __expf(-s[1][j][r]));
          const float gv = tanhf(s[2][j][r]);
          const float ov = 1.f / (1.f + __expf(-s[3][j][r]));
          const float cv = fv * cst[j][r] + iv * gv;
          cst[j][r] = cv;
          const float hv = ov * tanhf(cv);
          const int row = bm0 + r + 8 * half;
          out[((size_t)t * B_ + row) * H_ + coln] = hv;
          hbuf[row * H_ + coln] = hv;
          if (t == T_ - 1) {
            out[(size_t)T_ * B_ * H_ + (size_t)row * H_ + coln] = hv;   // hT
            out[(size_t)T_ * B_ * H_ + (size_t)B_ * H_ + (size_t)row * H_ + coln] = cv; // cT
          }
        }
      }
    }
    grid_barrier(bar, bar + 1);
  }
}

// ---------------------------------------------------------------------------
extern "C" void kernel_launch(void* const* d_in, const int* in_sizes, int n_in,
                              void* d_out, int out_size, void* d_ws, size_t ws_size,
                              hipStream_t stream) {
  const float* x   = (const float*)d_in[0];
  const float* Wf  = (const float*)d_in[1];
  const float* bf  = (const float*)d_in[2];
  const float* Wi  = (const float*)d_in[3];
  const float* bi  = (const float*)d_in[4];
  const float* Wg  = (const float*)d_in[5];
  const float* bg  = (const float*)d_in[6];
  const float* Wo  = (const float*)d_in[7];
  const float* bo  = (const float*)d_in[8];
  const float* Mf  = (const float*)d_in[9];
  const float* mbf = (const float*)d_in[10];
  const float* Mi  = (const float*)d_in[11];
  const float* mbi = (const float*)d_in[12];
  const float* Mg  = (const float*)d_in[13];
  const float* mbg = (const float*)d_in[14];
  const float* Mo  = (const float*)d_in[15];
  const float* mbo = (const float*)d_in[16];

  float* out  = (float*)d_out;
  float* xz   = (float*)d_ws;                        // [T*B][GC]  32 MB
  float* hbuf = xz + (size_t)T_ * B_ * GC_;          // [B][H]
  float* qbuf = hbuf + (size_t)B_ * H_;              // [B][GC]
  unsigned* bar = (unsigned*)(qbuf + (size_t)B_ * GC_);

  qlstm_init<<<1, 1, 0, stream>>>(bar);
  qlstm_xz<<<4096, 256, 0, stream>>>(x, Wf, bf, Wi, bi, Wg, bg, Wo, bo, xz);

  const size_t shmem =
      (size_t)(512 * WHP_ + 4 * 64 * MP_ + B_ * HP_) * sizeof(float); // 287,744 B
  qlstm_rec<<<NWG_, 256, shmem, stream>>>(Wf, Wi, Wg, Wo,
                                          Mf, mbf, Mi, mbi, Mg, mbg, Mo, mbo,
                                          xz, out, hbuf, qbuf, bar);
}